// PiCO_6073083757328
// MI455X (gfx1250) — compile-verified
//
#include <hip/hip_runtime.h>

typedef __attribute__((ext_vector_type(2))) float v2f;
typedef __attribute__((ext_vector_type(8))) float v8f;

#define BATCH   4096
#define DIN     1024
#define LOWD    128
#define NCLS    32
#define QLEN    65536

// ---- output offsets (in floats), tuple return order ----
#define O_OUTPUT    0u          // 4096*32
#define O_FEATURES  131072u     // 73728*128
#define O_PSEUDO    9568256u    // 73728*32
#define O_PARTIAL   11927552u   // 73728*32
#define O_SCOREPROT 14286848u   // 4096*32
#define O_PROTOS    14417920u   // 32*128
#define O_NEWQ      14422016u   // 65536*128
#define O_NEWQP     22810624u   // 65536*32
#define O_NEWQPT    24907776u   // 65536*32

// ---- workspace offsets (in floats) ----
#define W_CLS_T   0u        // 32 x 1024   (W_cls_q transposed)
#define W_PROJ_T  32768u    // 128 x 1024  (W_proj_q transposed)
#define W_PK_T    163840u   // 128 x 1024  (momentum key proj, transposed)
#define W_LABELS  294912u   // int[4096]
#define W_CHUNKA  299008u   // float[256]
#define W_CHUNKB  299264u   // float[256*128]

// ---------------------------------------------------------------------------
// Weight prep: transpose W_cls_q / W_proj_q; momentum-update + transpose key proj.
// ---------------------------------------------------------------------------
__global__ __launch_bounds__(256) void prep_weights(
    const float* __restrict__ Wcls, const float* __restrict__ Wproj,
    const float* __restrict__ WprojK, float* __restrict__ ws) {
  int idx = blockIdx.x * 256 + threadIdx.x;
  if (idx < 32768) {                       // WclsT[n*1024+k] = Wcls[k*32+n]
    int n = idx >> 10, k = idx & 1023;
    ws[W_CLS_T + idx] = Wcls[k * NCLS + n];
  } else if (idx < 163840) {               // WprojT[n*1024+k] = Wproj[k*128+n]
    int j = idx - 32768;
    int n = j >> 10, k = j & 1023;
    ws[W_PROJ_T + j] = Wproj[k * LOWD + n];
  } else if (idx < 294912) {               // WpkT = momentum(WprojK, Wproj)^T
    int j = idx - 163840;
    int n = j >> 10, k = j & 1023;
    ws[W_PK_T + j] = WprojK[k * LOWD + n] * 0.999f + Wproj[k * LOWD + n] * 0.001f;
  }
}

// ---------------------------------------------------------------------------
// WMMA f32 GEMM:  C[M x ldc] = A[M x K] @ BT[N x K]^T  (BT row n = column n of B)
// grid.x = M/16 row-tiles; one wave per 16-col tile (blockDim = 32 * N/16).
// ---------------------------------------------------------------------------
__global__ __launch_bounds__(256) void gemm_wmma_f32(
    const float* __restrict__ A, const float* __restrict__ BT,
    float* __restrict__ C, int K, int ldc) {
  const int wave = threadIdx.x >> 5;
  const int lane = threadIdx.x & 31;
  const int hf   = lane >> 4;     // lane half: 0 -> K even pair base 0, 1 -> base 2
  const int r    = lane & 15;
  const int m0   = blockIdx.x << 4;
  const int n0   = wave << 4;

  const float* aP = A  + (size_t)(m0 + r) * K + (hf << 1);
  const float* bP = BT + (size_t)(n0 + r) * K + (hf << 1);

  v8f acc = {0.f, 0.f, 0.f, 0.f, 0.f, 0.f, 0.f, 0.f};
  #pragma unroll 8
  for (int k0 = 0; k0 < K; k0 += 4) {
    v2f a = *(const v2f*)(aP + k0);   // A frag: VGPR v -> K = k0 + 2*hf + v, M = r
    v2f b = *(const v2f*)(bP + k0);   // B frag: VGPR v -> K = k0 + 2*hf + v, N = r
    acc = __builtin_amdgcn_wmma_f32_16x16x4_f32(false, a, false, b,
                                                (short)0, acc, false, false);
  }
  // C layout: VGPR i, lanes 0-15 -> M = i, lanes 16-31 -> M = i + 8; N = r
  const int row0 = m0 + (hf << 3);
  #pragma unroll
  for (int i = 0; i < 8; ++i)
    C[(size_t)(row0 + i) * ldc + n0 + r] = acc[i];
}

// ---------------------------------------------------------------------------
// Row L2-normalize (rows of 128): one wave per row, float4 (b128) per lane.
// ---------------------------------------------------------------------------
__global__ __launch_bounds__(256) void rownorm128(
    const float* __restrict__ in, float* __restrict__ out0,
    float* __restrict__ out1, int nrows) {
  int row  = blockIdx.x * 8 + (threadIdx.x >> 5);
  int lane = threadIdx.x & 31;
  if (row >= nrows) return;
  float4 v = ((const float4*)(in + (size_t)row * LOWD))[lane];
  float s = v.x * v.x + v.y * v.y + v.z * v.z + v.w * v.w;
  for (int off = 16; off >= 1; off >>= 1) s += __shfl_xor(s, off, 32);
  float inv = 1.0f / fmaxf(sqrtf(s), 1e-12f);
  v.x *= inv; v.y *= inv; v.z *= inv; v.w *= inv;
  ((float4*)(out0 + (size_t)row * LOWD))[lane] = v;
  if (out1) ((float4*)(out1 + (size_t)row * LOWD))[lane] = v;
}

// ---------------------------------------------------------------------------
// Softmax(logits+bias) * partial_Y, argmax -> labels, fan-out writes.
// One wave per row; lane == class (NCLS == warpSize == 32, perfect wave32 fit).
// ---------------------------------------------------------------------------
__global__ __launch_bounds__(256) void softmax_partial(
    float* __restrict__ logits, const float* __restrict__ bias,
    const float* __restrict__ partialY,
    float* __restrict__ pseudo, float* __restrict__ newqp,
    float* __restrict__ ptarget, float* __restrict__ newqpt,
    int* __restrict__ labels) {
  int row = blockIdx.x * 8 + (threadIdx.x >> 5);
  int c   = threadIdx.x & 31;
  size_t idx = (size_t)row * NCLS + c;
  float logit = logits[idx] + bias[c];
  logits[idx] = logit;                         // final "output" with bias
  float m = logit;
  for (int off = 16; off >= 1; off >>= 1) m = fmaxf(m, __shfl_xor(m, off, 32));
  float e = __expf(logit - m);
  float sum = e;
  for (int off = 16; off >= 1; off >>= 1) sum += __shfl_xor(sum, off, 32);
  float py = partialY[idx];
  float ps = (e / sum) * py;
  pseudo[idx]                       = ps;      // pseudo_scores rows 0..B
  pseudo[idx + (size_t)BATCH * NCLS] = ps;     // pseudo_scores rows B..2B
  newqp[idx]                        = ps;      // new_queue_pseudo rows 0..B
  ptarget[idx]                       = py;
  ptarget[idx + (size_t)BATCH * NCLS] = py;
  newqpt[idx]                        = py;
  // argmax, first-max-wins (lowest index on ties) to match jnp.argmax
  float v = ps; int bi = c;
  for (int off = 16; off >= 1; off >>= 1) {
    float ov = __shfl_xor(v, off, 32);
    int   oi = __shfl_xor(bi, off, 32);
    if (ov > v || (ov == v && oi < bi)) { v = ov; bi = oi; }
  }
  if (c == 0) labels[row] = bi;
}

// ---------------------------------------------------------------------------
// score_prot = softmax(q @ protos^T): protos staged transposed in LDS
// (conflict-free lane access), 8 rows per block, one wave per row.
// ---------------------------------------------------------------------------
__global__ __launch_bounds__(256) void score_prot_kernel(
    const float* __restrict__ qfeat, const float* __restrict__ protos,
    float* __restrict__ out) {
  __shared__ float pT[LOWD * NCLS];   // pT[k*32+n]
  __shared__ float qs[8][LOWD];
  int tid = threadIdx.x;
  for (int idx = tid; idx < NCLS * LOWD; idx += 256) {
    int n = idx >> 7, k = idx & 127;
    pT[k * NCLS + n] = protos[idx];
  }
  int row0 = blockIdx.x * 8;
  for (int idx = tid; idx < 8 * LOWD; idx += 256) {
    int rr = idx >> 7, k = idx & 127;
    qs[rr][k] = qfeat[(size_t)(row0 + rr) * LOWD + k];
  }
  __syncthreads();
  int wave = tid >> 5, c = tid & 31;
  float s = 0.f;
  #pragma unroll 4
  for (int k = 0; k < LOWD; ++k) s += qs[wave][k] * pT[k * NCLS + c];
  float m = s;
  for (int off = 16; off >= 1; off >>= 1) m = fmaxf(m, __shfl_xor(m, off, 32));
  float e = __expf(s - m);
  float sum = e;
  for (int off = 16; off >= 1; off >>= 1) sum += __shfl_xor(sum, off, 32);
  out[(size_t)(row0 + wave) * NCLS + c] = e / sum;
}

// ---------------------------------------------------------------------------
// Sequential prototype EMA as parallel affine-map composition.
// Phase 1: per (label, chunk) block of 128 dim-threads, fold 512 steps into
// (A, B): p -> A*p + B  (identity for non-matching labels).
// ---------------------------------------------------------------------------
__global__ __launch_bounds__(128) void proto_ema_chunks(
    const float* __restrict__ qfeat, const int* __restrict__ labels,
    float* __restrict__ chunkA, float* __restrict__ chunkB) {
  __shared__ int labs[512];
  int l = blockIdx.x >> 3;        // label
  int c = blockIdx.x & 7;         // chunk
  int t = threadIdx.x;            // dim
  int i0 = c * 512;
  for (int j = t; j < 512; j += 128) labs[j] = labels[i0 + j];
  __syncthreads();
  float A = 1.0f, Bc = 0.0f;
  for (int j = 0; j < 512; ++j) {
    float qv = qfeat[(size_t)(i0 + j) * LOWD + t];
    bool match = (labs[j] == l);
    A  = match ? A * 0.99f : A;
    Bc = match ? (0.99f * Bc + 0.01f * qv) : Bc;
  }
  chunkB[(size_t)blockIdx.x * LOWD + t] = Bc;
  if (t == 0) chunkA[blockIdx.x] = A;
}

// Phase 2: compose the 8 chunk maps in order, then L2-normalize prototypes.
__global__ __launch_bounds__(128) void proto_combine(
    const float* __restrict__ protos_in, const float* __restrict__ chunkA,
    const float* __restrict__ chunkB, float* __restrict__ protos_out) {
  __shared__ float P[NCLS * LOWD];
  int t = threadIdx.x;
  for (int l = 0; l < NCLS; ++l) {
    float p = protos_in[l * LOWD + t];
    #pragma unroll
    for (int c = 0; c < 8; ++c)
      p = chunkA[l * 8 + c] * p + chunkB[(size_t)(l * 8 + c) * LOWD + t];
    P[l * LOWD + t] = p;
  }
  __syncthreads();
  int wave = t >> 5, lane = t & 31;
  for (int l = wave; l < NCLS; l += 4) {
    float x0 = P[l*LOWD+lane],    x1 = P[l*LOWD+lane+32];
    float x2 = P[l*LOWD+lane+64], x3 = P[l*LOWD+lane+96];
    float s = x0*x0 + x1*x1 + x2*x2 + x3*x3;
    for (int off = 16; off >= 1; off >>= 1) s += __shfl_xor(s, off, 32);
    float inv = 1.0f / fmaxf(sqrtf(s), 1e-12f);
    protos_out[l*LOWD+lane]    = x0*inv;
    protos_out[l*LOWD+lane+32] = x1*inv;
    protos_out[l*LOWD+lane+64] = x2*inv;
    protos_out[l*LOWD+lane+96] = x3*inv;
  }
}

// ---------------------------------------------------------------------------
// Streaming float4 copy (b128 loads/stores) for concat tails & queue updates.
// ---------------------------------------------------------------------------
__global__ __launch_bounds__(256) void copy4(
    const float4* __restrict__ src, float4* __restrict__ dst, int n) {
  int i = blockIdx.x * 256 + threadIdx.x;
  if (i < n) dst[i] = src[i];
}

// ---------------------------------------------------------------------------
extern "C" void kernel_launch(void* const* d_in, const int* in_sizes, int n_in,
                              void* d_out, int out_size, void* d_ws, size_t ws_size,
                              hipStream_t stream) {
  (void)in_sizes; (void)n_in; (void)out_size; (void)ws_size;
  const float* img_q      = (const float*)d_in[0];
  const float* im_k       = (const float*)d_in[1];
  const float* partial_Y  = (const float*)d_in[2];
  const float* W_cls_q    = (const float*)d_in[3];
  const float* b_cls_q    = (const float*)d_in[4];
  const float* W_proj_q   = (const float*)d_in[5];
  // d_in[6] (W_cls_k) and d_in[7] (b_cls_k) feed only the discarded key logits.
  const float* W_proj_k   = (const float*)d_in[8];
  const float* prototypes = (const float*)d_in[9];
  const float* queue      = (const float*)d_in[10];
  const float* queue_ps   = (const float*)d_in[11];
  const float* queue_pt   = (const float*)d_in[12];

  float* out = (float*)d_out;
  float* wsf = (float*)d_ws;
  float* WclsT  = wsf + W_CLS_T;
  float* WprojT = wsf + W_PROJ_T;
  float* WpkT   = wsf + W_PK_T;
  int*   labels = (int*)(wsf + W_LABELS);
  float* chunkA = wsf + W_CHUNKA;
  float* chunkB = wsf + W_CHUNKB;

  // 1. weight transposes + momentum key-projection
  prep_weights<<<(294912 + 255) / 256, 256, 0, stream>>>(W_cls_q, W_proj_q, W_proj_k, wsf);

  // 2. GEMMs (WMMA f32 16x16x4)
  gemm_wmma_f32<<<BATCH / 16, 256, 0, stream>>>(img_q, WprojT,
      out + O_FEATURES, DIN, LOWD);                              // q_raw -> features[0:B]
  gemm_wmma_f32<<<BATCH / 16, 64, 0, stream>>>(img_q, WclsT,
      out + O_OUTPUT, DIN, NCLS);                                // raw logits
  gemm_wmma_f32<<<BATCH / 16, 256, 0, stream>>>(im_k, WpkT,
      out + O_FEATURES + (size_t)BATCH * LOWD, DIN, LOWD);       // k_raw -> features[B:2B]

  // 3. row L2-norms (q in place; k in place + into new_queue[0:B])
  rownorm128<<<BATCH / 8, 256, 0, stream>>>(out + O_FEATURES,
      out + O_FEATURES, nullptr, BATCH);
  rownorm128<<<BATCH / 8, 256, 0, stream>>>(out + O_FEATURES + (size_t)BATCH * LOWD,
      out + O_FEATURES + (size_t)BATCH * LOWD, out + O_NEWQ, BATCH);

  // 4. softmax*partial_Y, bias add, argmax labels, fan-out writes
  softmax_partial<<<BATCH / 8, 256, 0, stream>>>(out + O_OUTPUT, b_cls_q, partial_Y,
      out + O_PSEUDO, out + O_NEWQP, out + O_PARTIAL, out + O_NEWQPT, labels);

  // 5. score_prot = softmax(q @ protos^T)
  score_prot_kernel<<<BATCH / 8, 256, 0, stream>>>(out + O_FEATURES, prototypes,
      out + O_SCOREPROT);

  // 6. prototype EMA scan (parallel affine composition) + normalize
  proto_ema_chunks<<<NCLS * 8, 128, 0, stream>>>(out + O_FEATURES, labels, chunkA, chunkB);
  proto_combine<<<1, 128, 0, stream>>>(prototypes, chunkA, chunkB, out + O_PROTOS);

  // 7. concat tails + queue state updates (pure streaming copies)
  {
    int n;
    n = QLEN * LOWD / 4;                // features[2B:] = queue
    copy4<<<(n + 255) / 256, 256, 0, stream>>>((const float4*)queue,
        (float4*)(out + O_FEATURES + (size_t)2 * BATCH * LOWD), n);
    n = (QLEN - BATCH) * LOWD / 4;      // new_queue[B:] = queue[B:]
    copy4<<<(n + 255) / 256, 256, 0, stream>>>((const float4*)(queue + (size_t)BATCH * LOWD),
        (float4*)(out + O_NEWQ + (size_t)BATCH * LOWD), n);
    n = QLEN * NCLS / 4;                // pseudo_scores[2B:] = queue_pseudo
    copy4<<<(n + 255) / 256, 256, 0, stream>>>((const float4*)queue_ps,
        (float4*)(out + O_PSEUDO + (size_t)2 * BATCH * NCLS), n);
    n = (QLEN - BATCH) * NCLS / 4;      // new_queue_pseudo[B:] = queue_pseudo[B:]
    copy4<<<(n + 255) / 256, 256, 0, stream>>>((const float4*)(queue_ps + (size_t)BATCH * NCLS),
        (float4*)(out + O_NEWQP + (size_t)BATCH * NCLS), n);
    n = QLEN * NCLS / 4;                // partial_target[2B:] = queue_partial
    copy4<<<(n + 255) / 256, 256, 0, stream>>>((const float4*)queue_pt,
        (float4*)(out + O_PARTIAL + (size_t)2 * BATCH * NCLS), n);
    n = (QLEN - BATCH) * NCLS / 4;      // new_queue_partial[B:] = queue_partial[B:]
    copy4<<<(n + 255) / 256, 256, 0, stream>>>((const float4*)(queue_pt + (size_t)BATCH * NCLS),
        (float4*)(out + O_NEWQPT + (size_t)BATCH * NCLS), n);
  }
}